// QuantumLayer_15771119911266
// MI455X (gfx1250) — compile-verified
//
#include <hip/hip_runtime.h>

#define NQ 12
#define DIM 4096
#define NLAYERS 4

typedef __attribute__((ext_vector_type(2))) float v2f;
typedef __attribute__((ext_vector_type(8))) float v8f;

__device__ __forceinline__ v8f wmma4(v2f a, v2f b, v8f c) {
  // D(16x16,f32) += A(16x4,f32) * B(4x16,f32)
  return __builtin_amdgcn_wmma_f32_16x16x4_f32(false, a, false, b, (short)0, c,
                                               false, false);
}

// Composed map of the 12-CNOT ring (applied in reverse gate order), so that
// s_new[j] = s_old[cnot_map(j, off)].  GF(2)-linear in j.
__device__ __forceinline__ int cnot_map(int v, int off) {
#pragma unroll
  for (int g = NQ - 1; g >= 0; --g) {
    int t = g + off; if (t >= NQ) t -= NQ;
    int pc = NQ - 1 - g, pt = NQ - 1 - t;
    v ^= ((v >> pc) & 1) << pt;
  }
  return v;
}

__global__ __launch_bounds__(256) void qsim(const float* __restrict__ x,
                                            const float* __restrict__ w,
                                            float* __restrict__ out) {
  __shared__ float sr[DIM];        // state real
  __shared__ float si[DIM];        // state imag
  __shared__ float kr[3][256];     // 16x16 Kron unitary per 4-qubit group (re)
  __shared__ float ki[3][256];     // (im)
  __shared__ float g2r[NQ][4];     // current-round 2x2 gates (re)
  __shared__ float g2i[NQ][4];     // (im)
  __shared__ float red[NQ];        // output reduction

  const int b     = blockIdx.x;
  const int tid   = threadIdx.x;
  const int lane  = tid & 31;
  const int wv    = tid >> 5;             // wave id 0..7
  const int p16   = lane & 15;            // M (A) or N (B/D) coordinate
  const int koff  = (lane >> 4) << 1;     // K offset for A/B fragments
  const int mbase = (lane >> 4) << 3;     // M base for D fragment

  if (tid == 0) { __builtin_prefetch(x + b * NQ, 0, 0); __builtin_prefetch(w, 0, 0); }

  // init |00..0>
#pragma unroll
  for (int q = 0; q < 16; ++q) {
    int k = tid + (q << 8);
    sr[k] = (k == 0) ? 1.0f : 0.0f;
    si[k] = 0.0f;
  }
  if (tid < NQ) red[tid] = 0.0f;

  for (int round = 0; round <= NLAYERS; ++round) {
    // ---- build this round's 2x2 gate matrices -----------------------------
    if (tid < NQ) {
      if (round == 0) {
        float half = 1.57079632679489662f * x[b * NQ + tid];  // pi*x/2
        float s, c; sincosf(half, &s, &c);
        g2r[tid][0] = c;   g2r[tid][1] = -s;  g2r[tid][2] = s;   g2r[tid][3] = c;
        g2i[tid][0] = 0.f; g2i[tid][1] = 0.f; g2i[tid][2] = 0.f; g2i[tid][3] = 0.f;
      } else {
        const float* ww = w + ((round - 1) * NQ + tid) * 3;
        float phi = ww[0], th = ww[1], om = ww[2];
        float st, ct; sincosf(0.5f * th, &st, &ct);
        float sp, cp; sincosf(0.5f * (phi + om), &sp, &cp);
        float sm, cm; sincosf(0.5f * (phi - om), &sm, &cm);
        g2r[tid][0] =  cp * ct; g2i[tid][0] = -sp * ct;   // m00 = e^{-i(phi+om)/2} c
        g2r[tid][1] = -cm * st; g2i[tid][1] = -sm * st;   // m01 = -e^{ i(phi-om)/2} s
        g2r[tid][2] =  cm * st; g2i[tid][2] = -sm * st;   // m10 =  e^{-i(phi-om)/2} s
        g2r[tid][3] =  cp * ct; g2i[tid][3] =  sp * ct;   // m11 =  e^{ i(phi+om)/2} c
      }
    }
    __syncthreads();

    // ---- Kron-fuse 4 gates -> 16x16 unitary per group (one entry/thread) --
    {
      int r = tid >> 4, c = tid & 15;
#pragma unroll
      for (int g = 0; g < 3; ++g) {
        float pr = 1.f, pi = 0.f;
#pragma unroll
        for (int j = 0; j < 4; ++j) {
          int e = (((r >> (3 - j)) & 1) << 1) | ((c >> (3 - j)) & 1);
          float er = g2r[4 * g + j][e], ei = g2i[4 * g + j][e];
          float nr = pr * er - pi * ei;
          pi = pr * ei + pi * er;
          pr = nr;
        }
        kr[g][tid] = pr; ki[g][tid] = pi;
      }
    }
    __syncthreads();

    // ---- groups 0,1: new = U16 x S   (state is the B matrix) --------------
#pragma unroll
    for (int grp = 0; grp < 2; ++grp) {
      const int rs = (grp == 0) ? 256 : 16;   // stride of contracted index
      const int ts = (grp == 0) ? 16 : 256;   // stride of tile index
      v2f Ur[4], Uiv[4], Un[4];
#pragma unroll
      for (int kc = 0; kc < 4; ++kc) {        // A = U: element (m=p16, k)
        int k0 = 4 * kc + koff;
        v2f rr, ii;
        rr.x = kr[grp][p16 * 16 + k0]; rr.y = kr[grp][p16 * 16 + k0 + 1];
        ii.x = ki[grp][p16 * 16 + k0]; ii.y = ki[grp][p16 * 16 + k0 + 1];
        Ur[kc] = rr; Uiv[kc] = ii; Un[kc] = -ii;
      }
      v2f Br[2][4], Bi[2][4];
#pragma unroll
      for (int tt = 0; tt < 2; ++tt) {        // B = state tile: element (k, n=p16)
        int t = 2 * wv + tt;
#pragma unroll
        for (int kc = 0; kc < 4; ++kc) {
          int k0 = 4 * kc + koff;
          int i0 = k0 * rs + t * ts + p16;
          v2f rr, ii;
          rr.x = sr[i0]; rr.y = sr[i0 + rs];
          ii.x = si[i0]; ii.y = si[i0 + rs];
          Br[tt][kc] = rr; Bi[tt][kc] = ii;
        }
      }
      __syncthreads();                        // all source reads done -> in-place safe
#pragma unroll
      for (int tt = 0; tt < 2; ++tt) {
        int t = 2 * wv + tt;
        v8f Dr = {}, Di = {};
#pragma unroll
        for (int kc = 0; kc < 4; ++kc) Dr = wmma4(Ur[kc], Br[tt][kc], Dr);
#pragma unroll
        for (int kc = 0; kc < 4; ++kc) Dr = wmma4(Un[kc], Bi[tt][kc], Dr);
#pragma unroll
        for (int kc = 0; kc < 4; ++kc) Di = wmma4(Ur[kc], Bi[tt][kc], Di);
#pragma unroll
        for (int kc = 0; kc < 4; ++kc) Di = wmma4(Uiv[kc], Br[tt][kc], Di);
#pragma unroll
        for (int v = 0; v < 8; ++v) {
          int idx = (mbase + v) * rs + t * ts + p16;
          sr[idx] = Dr[v]; si[idx] = Di[v];
        }
      }
      __syncthreads();
    }

    // ---- group 2: new = S x U16^T   (state is the A matrix) ---------------
    {
      v2f Btr[4], Bti[4], Btn[4];
#pragma unroll
      for (int kc = 0; kc < 4; ++kc) {        // B = U^T: element (k=c, n=r=p16)
        int k0 = 4 * kc + koff;
        v2f rr, ii;
        rr.x = kr[2][p16 * 16 + k0]; rr.y = kr[2][p16 * 16 + k0 + 1];
        ii.x = ki[2][p16 * 16 + k0]; ii.y = ki[2][p16 * 16 + k0 + 1];
        Btr[kc] = rr; Bti[kc] = ii; Btn[kc] = -ii;
      }
      v2f Ar[2][4], Aiv[2][4], An[2][4];
#pragma unroll
      for (int tt = 0; tt < 2; ++tt) {        // A = state rows: element (m=p16, k)
        int t = 2 * wv + tt;
#pragma unroll
        for (int kc = 0; kc < 4; ++kc) {
          int k0 = 4 * kc + koff;
          int i0 = (t << 8) + (p16 << 4) + k0;
          v2f rr, ii;
          rr.x = sr[i0]; rr.y = sr[i0 + 1];
          ii.x = si[i0]; ii.y = si[i0 + 1];
          Ar[tt][kc] = rr; Aiv[tt][kc] = ii; An[tt][kc] = -ii;
        }
      }
      __syncthreads();
#pragma unroll
      for (int tt = 0; tt < 2; ++tt) {
        int t = 2 * wv + tt;
        v8f Dr = {}, Di = {};
#pragma unroll
        for (int kc = 0; kc < 4; ++kc) Dr = wmma4(Ar[tt][kc], Btr[kc], Dr);
#pragma unroll
        for (int kc = 0; kc < 4; ++kc) Dr = wmma4(An[tt][kc], Bti[kc], Dr);
#pragma unroll
        for (int kc = 0; kc < 4; ++kc) Di = wmma4(Ar[tt][kc], Bti[kc], Di);
#pragma unroll
        for (int kc = 0; kc < 4; ++kc) Di = wmma4(Aiv[tt][kc], Btr[kc], Di);
#pragma unroll
        for (int v = 0; v < 8; ++v) {
          int idx = (t << 8) + ((mbase + v) << 4) + p16;
          sr[idx] = Dr[v]; si[idx] = Di[v];
        }
      }
      __syncthreads();
    }

    // ---- CNOT ring: composed basis permutation, in-place gather -----------
    // T is GF(2)-linear: T(tid | q<<8) = T(tid) ^ T(q<<8).  T(tid) is one
    // per-thread VALU chain; the q-masks are wave-uniform -> SALU.
    {
      const int off = (round == 0) ? 1 : round;  // r = 1,1,2,3,4
      const int tb = cnot_map(tid, off);         // per-thread (VALU)
      int mH[4];                                 // uniform basis masks (SALU)
#pragma unroll
      for (int e = 0; e < 4; ++e) mH[e] = cnot_map(256 << e, off);
      float vr[16], vi[16];
#pragma unroll
      for (int q = 0; q < 16; ++q) {
        int h = 0;                               // uniform per q (SALU)
        if (q & 1) h ^= mH[0];
        if (q & 2) h ^= mH[1];
        if (q & 4) h ^= mH[2];
        if (q & 8) h ^= mH[3];
        int s = tb ^ h;                          // one v_xor per amplitude
        vr[q] = sr[s]; vi[q] = si[s];
      }
      __syncthreads();
#pragma unroll
      for (int q = 0; q < 16; ++q) {
        int j = tid + (q << 8);
        sr[j] = vr[q]; si[j] = vi[q];
      }
      __syncthreads();
    }
  }

  // ---- <Z_i> expectations --------------------------------------------------
  float acc[NQ];
#pragma unroll
  for (int i = 0; i < NQ; ++i) acc[i] = 0.f;
#pragma unroll
  for (int q = 0; q < 16; ++q) {
    int k = tid + (q << 8);
    float re = sr[k], im = si[k];
    float pp = re * re + im * im;
#pragma unroll
    for (int i = 0; i < NQ; ++i)
      acc[i] += ((k >> (NQ - 1 - i)) & 1) ? -pp : pp;  // i<4: folds to add/sub
  }
#pragma unroll
  for (int i = 0; i < NQ; ++i) {
    float v = acc[i];
#pragma unroll
    for (int d = 16; d >= 1; d >>= 1) v += __shfl_xor(v, d, 32);
    if (lane == 0) atomicAdd(&red[i], v);   // ds_add_f32
  }
  __syncthreads();
  if (tid < NQ) out[b * NQ + tid] = red[tid];
}

extern "C" void kernel_launch(void* const* d_in, const int* in_sizes, int n_in,
                              void* d_out, int out_size, void* d_ws, size_t ws_size,
                              hipStream_t stream) {
  const float* x = (const float*)d_in[0];     // (1024, 12)
  const float* w = (const float*)d_in[1];     // (4, 12, 3)
  float* out = (float*)d_out;                 // (1024, 12)
  int batch = in_sizes[0] / NQ;
  qsim<<<batch, 256, 0, stream>>>(x, w, out);
}